// AC_MSA_81398220194002
// MI455X (gfx1250) — compile-verified
//
#include <hip/hip_runtime.h>
#include <math.h>

#define NUM_HEADS 4
#define HEAD_DIM  48
#define DIM       192
#define CAT       64
#define BATCH     2
#define NTOK      65000
#define GS        128
#define NGRP      508          // ceil(65000/128)
#define NPAD      (NGRP*GS)    // 65024
#define PADN      (NPAD-NTOK)  // 24
#define SORT_BLK  256
#define NSBLK     ((NTOK + SORT_BLK - 1)/SORT_BLK)  // 254
#define LDS_STRIDE 50

typedef float v2f __attribute__((ext_vector_type(2)));
typedef float v8f __attribute__((ext_vector_type(8)));

__device__ inline v8f wmma_f32(v2f a, v2f b, v8f c) {
  // D = A(16x4, f32) x B(4x16, f32) + C(16x16, f32)
  return __builtin_amdgcn_wmma_f32_16x16x4_f32(false, a, false, b, (short)0, c, false, false);
}

__device__ inline v8f v8f_zero() {
  v8f z;
#pragma unroll
  for (int i = 0; i < 8; ++i) z[i] = 0.0f;
  return z;
}

// ---------------- Kernel A: per-token argmax + per-block category histogram ----------------
__global__ void k_argmax_hist(const float* __restrict__ sim, int* __restrict__ tkid,
                              int* __restrict__ hist) {
  __shared__ int lh[CAT];
  const int b = blockIdx.y, blk = blockIdx.x, tid = threadIdx.x;
  if (tid < CAT) lh[tid] = 0;
  __syncthreads();
  const int i = blk * SORT_BLK + tid;
  int cat = -1;
  if (i < NTOK) {
    const float* s = sim + ((size_t)b * NTOK + i) * CAT;
    float best = s[0]; cat = 0;
#pragma unroll 4
    for (int c = 1; c < CAT; ++c) { float v = s[c]; if (v > best) { best = v; cat = c; } }
    tkid[(size_t)b * NTOK + i] = cat;
    atomicAdd(&lh[cat], 1);
  }
  __syncthreads();
  if (tid < CAT) hist[((size_t)b * CAT + tid) * NSBLK + blk] = lh[tid];
}

// ---------------- Kernel B: exclusive scan of histograms -> scatter offsets ----------------
__global__ void k_scan(int* __restrict__ hist) {
  __shared__ int tot[BATCH * CAT];
  __shared__ int base[BATCH * CAT];
  const int t = threadIdx.x;  // 0..127 => (batch, cat)
  if (t < BATCH * CAT) {
    int* h = hist + (size_t)t * NSBLK;
    int run = 0;
    for (int j = 0; j < NSBLK; ++j) { int v = h[j]; h[j] = run; run += v; }
    tot[t] = run;
  }
  __syncthreads();
  if (t < BATCH) {
    int run = 0;
    for (int c = 0; c < CAT; ++c) { base[t * CAT + c] = run; run += tot[t * CAT + c]; }
  }
  __syncthreads();
  if (t < BATCH * CAT) {
    int* h = hist + (size_t)t * NSBLK;
    const int bs = base[t];
    for (int j = 0; j < NSBLK; ++j) h[j] += bs;
  }
}

// ---------------- Kernel C: stable rank within block + scatter of indices ----------------
__global__ void k_scatter(const int* __restrict__ tkid, const int* __restrict__ hist,
                          int* __restrict__ sortIdx) {
  __shared__ int cats[SORT_BLK];
  const int b = blockIdx.y, blk = blockIdx.x, tid = threadIdx.x;
  const int i = blk * SORT_BLK + tid;
  const int cat = (i < NTOK) ? tkid[(size_t)b * NTOK + i] : -1;
  cats[tid] = cat;
  __syncthreads();
  if (i < NTOK) {
    int rank = 0;
    for (int j = 0; j < tid; ++j) rank += (cats[j] == cat) ? 1 : 0;
    const int pos = hist[((size_t)b * CAT + cat) * NSBLK + blk] + rank;
    sortIdx[(size_t)b * NPAD + pos] = i;
  }
}

// ---------------- Kernel D: mirrored padding of the sorted index tail ----------------
__global__ void k_pad(int* __restrict__ sortIdx) {
  const int j = threadIdx.x, b = blockIdx.x;
  if (j < PADN)
    sortIdx[(size_t)b * NPAD + NTOK + j] = sortIdx[(size_t)b * NPAD + NTOK - 1 - j];
}

// ---------------- Kernel E: grouped attention with fp32 WMMA ----------------
// grid (NGRP, NUM_HEADS, BATCH), 256 threads (8 waves). Each wave owns 16 query columns.
// Computes logits^T = K * Q^T  (lane = query column), softmax in registers,
// then OUT^T = V^T * attn^T.
__global__ __launch_bounds__(256) void k_attn(const float* __restrict__ qkv,
                                              const int* __restrict__ sortIdx,
                                              const float* __restrict__ logit_scale,
                                              float* __restrict__ attnOut) {
  __shared__ float kl[GS * LDS_STRIDE];
  __shared__ float vl[GS * LDS_STRIDE];
  const int g = blockIdx.x, h = blockIdx.y, b = blockIdx.z;
  const int tid = threadIdx.x;
  const int lane = tid & 31;
  const int w = tid >> 5;       // wave 0..7
  const int laneL = lane & 15;
  const int hi = lane >> 4;     // half-wave selector

  const int* sIdx = sortIdx + (size_t)b * NPAD + (size_t)g * GS;
  const float* qkvB = qkv + (size_t)b * NTOK * (3 * DIM);

  // Cooperative gather of K, V for this (group, head) into LDS.
  for (int e = tid; e < GS * HEAD_DIM; e += 256) {
    const int t = e / HEAD_DIM, d = e - t * HEAD_DIM;
    const int tok = sIdx[t];
    const float* src = qkvB + (size_t)tok * (3 * DIM) + h * HEAD_DIM + d;
    kl[t * LDS_STRIDE + d] = src[DIM];      // K
    vl[t * LDS_STRIDE + d] = src[2 * DIM];  // V
  }

  // Q fragments for this wave's 16 queries (B operand of K*Q^T), gathered from global.
  const int q    = w * 16 + laneL;   // query row within group
  const int qpos = g * GS + q;       // sorted position
  const int qtok = sIdx[q];          // original token
  const float* qp = qkvB + (size_t)qtok * (3 * DIM) + h * HEAD_DIM + 2 * hi;
  v2f qreg[12];
#pragma unroll
  for (int s = 0; s < 12; ++s) { qreg[s].x = qp[4 * s]; qreg[s].y = qp[4 * s + 1]; }
  __syncthreads();

  const float ls = logit_scale[0];
  const float scale = expf(fminf(ls, 4.60517018598809f));  // exp(min(ls, ln 100))

  // logits^T accumulators: 8 key-tiles x (16 keys x 16 queries)
  v8f acc[8];
#pragma unroll
  for (int t = 0; t < 8; ++t) acc[t] = v8f_zero();

#pragma unroll
  for (int kt = 0; kt < 8; ++kt) {
    v8f c = acc[kt];
#pragma unroll
    for (int s = 0; s < 12; ++s) {
      v2f a;  // A = K tile: row = key, K-dim = feature
      const float* kp = &kl[(kt * 16 + laneL) * LDS_STRIDE + 4 * s + 2 * hi];
      a.x = kp[0]; a.y = kp[1];
      c = wmma_f32(a, qreg[s], c);
    }
    acc[kt] = c;
  }

  // Softmax over keys (fixed query per lane-column; partner lane holds the other 64 keys).
  float m = -INFINITY;
#pragma unroll
  for (int t = 0; t < 8; ++t)
#pragma unroll
    for (int r = 0; r < 8; ++r) m = fmaxf(m, acc[t][r]);
  m = fmaxf(m, __shfl_xor(m, 16, 32));
  const float sm = scale * m;
  float sum = 0.0f;
#pragma unroll
  for (int t = 0; t < 8; ++t)
#pragma unroll
    for (int r = 0; r < 8; ++r) {
      const float e = expf(scale * acc[t][r] - sm);
      acc[t][r] = e;
      sum += e;
    }
  sum += __shfl_xor(sum, 16, 32);
  const float inv = 1.0f / sum;
#pragma unroll
  for (int t = 0; t < 8; ++t)
#pragma unroll
    for (int r = 0; r < 8; ++r) acc[t][r] *= inv;

  // OUT^T = V^T (48x128) * attn^T (128x16). B-fragment rebuilt from acc via xor-16 exchange.
  v8f outv[3];
#pragma unroll
  for (int dt = 0; dt < 3; ++dt) outv[dt] = v8f_zero();

#pragma unroll
  for (int s = 0; s < 32; ++s) {
    const int t  = s >> 2;         // key tile
    const int l0 = (4 * s) & 15;   // first local key of this step (0,4,8,12)
    const int e  = l0 & 7;
    v2f bfr;
    if (l0 < 8) {
      const float p2 = __shfl_xor(acc[t][e + 2], 16, 32);
      const float p3 = __shfl_xor(acc[t][e + 3], 16, 32);
      bfr.x = hi ? p2 : acc[t][e];
      bfr.y = hi ? p3 : acc[t][e + 1];
    } else {
      const float p0 = __shfl_xor(acc[t][e], 16, 32);
      const float p1 = __shfl_xor(acc[t][e + 1], 16, 32);
      bfr.x = hi ? acc[t][e + 2] : p0;
      bfr.y = hi ? acc[t][e + 3] : p1;
    }
    const int krow = 4 * s + 2 * hi;
#pragma unroll
    for (int dt = 0; dt < 3; ++dt) {
      v2f a;  // A = V^T tile: row = dim, K-dim = key
      a.x = vl[(krow)     * LDS_STRIDE + dt * 16 + laneL];
      a.y = vl[(krow + 1) * LDS_STRIDE + dt * 16 + laneL];
      outv[dt] = wmma_f32(a, bfr, outv[dt]);
    }
  }

  // Scatter back to original token order (skip padded queries).
  if (qpos < NTOK) {
    float* op = attnOut + ((size_t)b * NTOK + qtok) * DIM + h * HEAD_DIM + 8 * hi;
#pragma unroll
    for (int dt = 0; dt < 3; ++dt)
#pragma unroll
      for (int r = 0; r < 8; ++r) op[dt * 16 + r] = outv[dt][r];
  }
}

// ---------------- Kernel F: output projection  out = X @ W^T + b  with fp32 WMMA ----------------
// grid (ceil(NTOK/128), BATCH), 256 threads (8 waves). Each wave: 16 token rows x 192 outputs.
__global__ __launch_bounds__(256) void k_proj(const float* __restrict__ x,
                                              const float* __restrict__ w,
                                              const float* __restrict__ bias,
                                              float* __restrict__ out) {
  __shared__ float wl[16 * DIM];  // W^T chunk: 16 input-features x 192 outputs (12 KB)
  const int b = blockIdx.y;
  const int tokBase = blockIdx.x * 128;
  const int tid = threadIdx.x;
  const int lane = tid & 31, wv = tid >> 5, laneL = lane & 15, hi = lane >> 4;
  const int rowBase = tokBase + wv * 16;

  v8f acc[12];
#pragma unroll
  for (int nt = 0; nt < 12; ++nt) acc[nt] = v8f_zero();

  const int tokA = rowBase + laneL;
  const int tokC = (tokA < NTOK) ? tokA : (NTOK - 1);
  const float* xp = x + ((size_t)b * NTOK + tokC) * DIM + 2 * hi;

  for (int ch = 0; ch < 12; ++ch) {  // 12 chunks of 16 input features
    __syncthreads();
    for (int e = tid; e < 16 * DIM; e += 256) {
      const int cp = e / DIM, o = e - cp * DIM;
      wl[cp * DIM + o] = w[(size_t)o * DIM + ch * 16 + cp];  // W^T[c][o]
    }
    __syncthreads();
    v2f afr[4];
#pragma unroll
    for (int s2 = 0; s2 < 4; ++s2) {
      afr[s2].x = xp[ch * 16 + 4 * s2];
      afr[s2].y = xp[ch * 16 + 4 * s2 + 1];
    }
#pragma unroll
    for (int s2 = 0; s2 < 4; ++s2) {
#pragma unroll
      for (int nt = 0; nt < 12; ++nt) {
        v2f bf;
        bf.x = wl[(4 * s2 + 2 * hi)     * DIM + nt * 16 + laneL];
        bf.y = wl[(4 * s2 + 2 * hi + 1) * DIM + nt * 16 + laneL];
        acc[nt] = wmma_f32(afr[s2], bf, acc[nt]);
      }
    }
  }

#pragma unroll
  for (int nt = 0; nt < 12; ++nt) {
    const int o = nt * 16 + laneL;
    const float pb = bias[o];
#pragma unroll
    for (int r = 0; r < 8; ++r) {
      const int tok = rowBase + r + 8 * hi;
      if (tok < NTOK) out[((size_t)b * NTOK + tok) * DIM + o] = acc[nt][r] + pb;
    }
  }
}

// ---------------- Host launcher ----------------
extern "C" void kernel_launch(void* const* d_in, const int* in_sizes, int n_in,
                              void* d_out, int out_size, void* d_ws, size_t ws_size,
                              hipStream_t stream) {
  const float* qkv         = (const float*)d_in[0];
  const float* sim         = (const float*)d_in[1];
  const float* proj_w      = (const float*)d_in[2];
  const float* proj_b      = (const float*)d_in[3];
  const float* logit_scale = (const float*)d_in[4];
  // d_in[5] = x_size (65000), baked into constants above.
  float* out = (float*)d_out;

  char* ws = (char*)d_ws;
  float* attnOut = (float*)ws;
  size_t off = (size_t)BATCH * NTOK * DIM * sizeof(float);
  int* tkid = (int*)(ws + off);    off += (size_t)BATCH * NTOK * sizeof(int);
  int* hist = (int*)(ws + off);    off += (size_t)BATCH * CAT * NSBLK * sizeof(int);
  int* sortIdx = (int*)(ws + off); off += (size_t)BATCH * NPAD * sizeof(int);
  (void)ws_size; (void)in_sizes; (void)n_in; (void)out_size;

  dim3 gSort(NSBLK, BATCH);
  k_argmax_hist<<<gSort, SORT_BLK, 0, stream>>>(sim, tkid, hist);
  k_scan<<<1, 128, 0, stream>>>(hist);
  k_scatter<<<gSort, SORT_BLK, 0, stream>>>(tkid, hist, sortIdx);
  k_pad<<<BATCH, 32, 0, stream>>>(sortIdx);

  dim3 gAttn(NGRP, NUM_HEADS, BATCH);
  k_attn<<<gAttn, 256, 0, stream>>>(qkv, sortIdx, logit_scale, attnOut);

  dim3 gProj((NTOK + 127) / 128, BATCH);
  k_proj<<<gProj, 256, 0, stream>>>(attnOut, proj_w, proj_b, out);
}